// DynamicEmbedder_4D_pc0_2087354106092
// MI455X (gfx1250) — compile-verified
//
#include <hip/hip_runtime.h>
#include <hip/hip_bf16.h>

#define GX 256
#define GY 256
#define GZ 16
#define NUM_VOX (GX * GY * GZ)
#define CFEAT 32
#define BB 2
#define NPTS 150000
#define BN_EPS 1e-3f
#define VS 0.4f
#define INV_VS 2.5f
#define LOX (-51.2f)
#define LOY (-51.2f)
#define LOZ (-3.2f)

typedef __attribute__((ext_vector_type(2))) float v2f;
typedef __attribute__((ext_vector_type(8))) float v8f;

__device__ __forceinline__ void voxel_idx(float x, float y, float z,
                                          int& ix, int& iy, int& iz) {
    ix = (int)floorf((x - LOX) * INV_VS);
    iy = (int)floorf((y - LOY) * INV_VS);
    iz = (int)floorf((z - LOZ) * INV_VS);
    ix = min(max(ix, 0), GX - 1);
    iy = min(max(iy, 0), GY - 1);
    iz = min(max(iz, 0), GZ - 1);
}

__device__ __forceinline__ void build_feats(float x, float y, float z,
                                            const float* __restrict__ sums4,
                                            int ix, int iy, int iz,
                                            float f[9]) {
    float cnt = fmaxf(sums4[3], 1.0f);
    float inv = 1.0f / cnt;
    float mx = sums4[0] * inv, my = sums4[1] * inv, mz = sums4[2] * inv;
    float cx = (float)ix * VS + LOX + 0.5f * VS;
    float cy = (float)iy * VS + LOY + 0.5f * VS;
    float cz = (float)iz * VS + LOZ + 0.5f * VS;
    f[0] = x; f[1] = y; f[2] = z;
    f[3] = x - mx; f[4] = y - my; f[5] = z - mz;
    f[6] = x - cx; f[7] = y - cy; f[8] = z - cz;
}

// ---------------- zero fill (float4 grid-stride) ----------------
__global__ void zero_f4(float4* __restrict__ p, long n4) {
    long stride = (long)gridDim.x * blockDim.x;
    for (long i = (long)blockIdx.x * blockDim.x + threadIdx.x; i < n4; i += stride)
        p[i] = make_float4(0.f, 0.f, 0.f, 0.f);
}

// ---------------- pass 1: voxel scatter of [xyz,1] ----------------
__global__ void __launch_bounds__(256)
scatter_sums(const float* __restrict__ pc, int* __restrict__ seg_ws,
             float* __restrict__ sums) {
    long t = (long)blockIdx.x * blockDim.x + threadIdx.x;
    if (t >= (long)BB * NPTS) return;
    int b = (int)(t / NPTS);
    float x = pc[t * 3 + 0], y = pc[t * 3 + 1], z = pc[t * 3 + 2];
    int ix, iy, iz;
    voxel_idx(x, y, z, ix, iy, iz);
    int sg = (iz * GY + iy) * GX + ix;
    seg_ws[t] = sg;
    float* s = sums + ((long)b * NUM_VOX + sg) * 4;
    atomicAdd(s + 0, x);
    atomicAdd(s + 1, y);
    atomicAdd(s + 2, z);
    atomicAdd(s + 3, 1.0f);
}

// ---------------- pass 2: h = feats @ W^T via V_WMMA_F32_16X16X4_F32,
//                  accumulate per-channel BN sum / sumsq ----------------
__global__ void __launch_bounds__(256)
h_stats_wmma(const float* __restrict__ pc, const float* __restrict__ W,
             const int* __restrict__ seg_ws, const float* __restrict__ sums,
             float* __restrict__ bn_acc /* [B][2][32] : sum, sumsq */) {
    const int lane = threadIdx.x & 31;
    const int wave = threadIdx.x >> 5;
    const int tile = blockIdx.x * 8 + wave;      // 16-point tile id
    const int NT = NPTS / 16;                    // 9375 exact tiles
    if (tile >= NT) return;                      // wave-uniform exit: EXEC stays all-1
    const int b  = blockIdx.y;
    const int m  = lane & 15;                    // point row within tile / channel col
    const int hi = lane >> 4;

    const long p = (long)b * NPTS + (long)tile * 16 + m;
    const float x = pc[p * 3 + 0], y = pc[p * 3 + 1], z = pc[p * 3 + 2];
    int ix, iy, iz;
    voxel_idx(x, y, z, ix, iy, iz);
    const int sg = seg_ws[p];
    const float* s4 = sums + ((long)b * NUM_VOX + sg) * 4;

    float f[9];
    build_feats(x, y, z, s4, ix, iy, iz, f);

    v8f acc0 = {};   // channels 0..15
    v8f acc1 = {};   // channels 16..31

#pragma unroll
    for (int st = 0; st < 3; ++st) {
        // A fragment: row M=m, K = 4*st + {2*hi, 2*hi+1} (zero-pad K>=9)
        const int kb = 4 * st;
        float a0 = hi ? ((kb + 2 < 9) ? f[kb + 2] : 0.f) : f[kb + 0];
        float a1 = hi ? ((kb + 3 < 9) ? f[kb + 3] : 0.f)
                      : ((kb + 1 < 9) ? f[kb + 1] : 0.f);
        v2f a; a.x = a0; a.y = a1;
        // B fragments: B[k][n] = W[n][k], n = m (+16 for upper half of channels)
        const int k = kb + 2 * hi;
        v2f b0, b1;
        b0.x = (k     < 9) ? W[m * 9 + k]            : 0.f;
        b0.y = (k + 1 < 9) ? W[m * 9 + k + 1]        : 0.f;
        b1.x = (k     < 9) ? W[(m + 16) * 9 + k]     : 0.f;
        b1.y = (k + 1 < 9) ? W[(m + 16) * 9 + k + 1] : 0.f;

        acc0 = __builtin_amdgcn_wmma_f32_16x16x4_f32(
            false, a, false, b0, (short)0, acc0, false, false);
        acc1 = __builtin_amdgcn_wmma_f32_16x16x4_f32(
            false, a, false, b1, (short)0, acc1, false, false);
    }

    // D layout: VGPR r, this lane -> point (tile*16 + (hi ? r+8 : r)), channel m (+16).
    // Each lane reduces its 8 points for one fixed channel -> BN partials.
    float s0 = 0.f, q0 = 0.f, s1 = 0.f, q1 = 0.f;
#pragma unroll
    for (int r = 0; r < 8; ++r) {
        float v = acc0[r]; s0 += v; q0 += v * v;
        v = acc1[r];       s1 += v; q1 += v * v;
    }
    float* bn = bn_acc + b * 64;
    atomicAdd(bn + m,            s0);
    atomicAdd(bn + 32 + m,       q0);
    atomicAdd(bn + 16 + m,       s1);
    atomicAdd(bn + 32 + 16 + m,  q1);
}

// ---------------- pass 3: BN stats finalize ----------------
__global__ void bn_finalize(const float* __restrict__ bn_acc,
                            float* __restrict__ bn_stats /* [B][2][32]: mu, invstd */) {
    int t = threadIdx.x;
    if (t >= BB * 32) return;
    int b = t / 32, c = t % 32;
    float sum = bn_acc[b * 64 + c];
    float sq  = bn_acc[b * 64 + 32 + c];
    float mu  = sum * (1.0f / (float)NPTS);
    float var = sq * (1.0f / (float)NPTS) - mu * mu;
    bn_stats[b * 64 + c]      = mu;
    bn_stats[b * 64 + 32 + c] = rsqrtf(var + BN_EPS);
}

// ---------------- pass 4: point feats + voxel accumulate ----------------
__global__ void __launch_bounds__(256)
point_feats_kernel(const float* __restrict__ pc, const float* __restrict__ W,
                   const float* __restrict__ gamma, const float* __restrict__ beta,
                   const int* __restrict__ seg_ws, const float* __restrict__ sums,
                   const float* __restrict__ bn_stats,
                   float* __restrict__ pf_out, float* __restrict__ vox_out) {
    __shared__ float Ws[CFEAT * 9];
    for (int i = threadIdx.x; i < CFEAT * 9; i += blockDim.x) Ws[i] = W[i];
    __syncthreads();

    long t = (long)blockIdx.x * blockDim.x + threadIdx.x;
    if (t >= (long)BB * NPTS * 32) return;
    int  c  = (int)(t & 31);
    long bp = t >> 5;                 // b*NPTS + p
    int  b  = (int)(bp / NPTS);

    float x = pc[bp * 3 + 0], y = pc[bp * 3 + 1], z = pc[bp * 3 + 2];
    int ix, iy, iz;
    voxel_idx(x, y, z, ix, iy, iz);
    int sg = seg_ws[bp];
    const float* s4 = sums + ((long)b * NUM_VOX + sg) * 4;

    float f[9];
    build_feats(x, y, z, s4, ix, iy, iz, f);

    const float* w = Ws + c * 9;
    float h = 0.f;
#pragma unroll
    for (int k = 0; k < 9; ++k) h = fmaf(f[k], w[k], h);

    float mu = bn_stats[b * 64 + c];
    float is = bn_stats[b * 64 + 32 + c];
    float v  = fmaf((h - mu) * is, gamma[c], beta[c]);
    v = fmaxf(v, 0.0f);

    pf_out[t] = v;
    atomicAdd(&vox_out[((long)b * NUM_VOX + sg) * 32 + c], v);
}

// ---------------- pass 5: voxel mean (divide by count) ----------------
__global__ void __launch_bounds__(256)
vox_div(float* __restrict__ vox, const float* __restrict__ sums) {
    long t = (long)blockIdx.x * blockDim.x + threadIdx.x;
    if (t >= (long)BB * NUM_VOX * 8) return;    // float4 granules
    long bv = t >> 3;                           // b*NUM_VOX + v
    float cnt = sums[bv * 4 + 3];
    float inv = 1.0f / fmaxf(cnt, 1.0f);
    float4* p = (float4*)vox + t;
    float4 q = *p;
    q.x *= inv; q.y *= inv; q.z *= inv; q.w *= inv;
    *p = q;
}

extern "C" void kernel_launch(void* const* d_in, const int* in_sizes, int n_in,
                              void* d_out, int out_size, void* d_ws, size_t ws_size,
                              hipStream_t stream) {
    const float* pc    = (const float*)d_in[0];   // [B,N,3]
    const float* W     = (const float*)d_in[1];   // [32,9]
    const float* gamma = (const float*)d_in[2];   // [32]
    const float* beta  = (const float*)d_in[3];   // [32]

    float* out     = (float*)d_out;
    float* vox_out = out;                                   // [B,V,32]
    float* pf_out  = out + (long)BB * NUM_VOX * 32;         // [B,N,32]

    // workspace layout (~35 MB)
    char*  ws  = (char*)d_ws;
    size_t off = 0;
    int*   seg_ws = (int*)(ws + off);
    off += ((size_t)BB * NPTS * sizeof(int) + 255) & ~(size_t)255;
    float* sums = (float*)(ws + off);                       // [B,V,4]
    off += (size_t)BB * NUM_VOX * 4 * sizeof(float);
    float* bn_acc = (float*)(ws + off);                     // [B,2,32]
    off += 256;
    float* bn_stats = (float*)(ws + off);                   // [B,2,32]

    // zero the accumulators + voxel output region (re-done every call)
    long vox_n4  = (long)BB * NUM_VOX * 32 / 4;
    long sums_n4 = (long)BB * NUM_VOX * 4 / 4;
    zero_f4<<<4096, 256, 0, stream>>>((float4*)vox_out, vox_n4);
    zero_f4<<<2048, 256, 0, stream>>>((float4*)sums, sums_n4);
    zero_f4<<<1, 64, 0, stream>>>((float4*)bn_acc, (long)(BB * 64 / 4));

    long npts_total = (long)BB * NPTS;
    scatter_sums<<<(int)((npts_total + 255) / 256), 256, 0, stream>>>(pc, seg_ws, sums);

    dim3 g((NPTS / 16 + 7) / 8, BB);
    h_stats_wmma<<<g, 256, 0, stream>>>(pc, W, seg_ws, sums, bn_acc);

    bn_finalize<<<1, 64, 0, stream>>>(bn_acc, bn_stats);

    long pf_total = (long)BB * NPTS * 32;
    point_feats_kernel<<<(int)((pf_total + 255) / 256), 256, 0, stream>>>(
        pc, W, gamma, beta, seg_ws, sums, bn_stats, pf_out, vox_out);

    long vd_total = (long)BB * NUM_VOX * 8;
    vox_div<<<(int)((vd_total + 255) / 256), 256, 0, stream>>>(vox_out, sums);
}